// MultiHeadAttention_17798344474903
// MI455X (gfx1250) — compile-verified
//
#include <hip/hip_runtime.h>
#include <hip/hip_bf16.h>

// Problem constants (match reference)
#define N_TOT   16384      // BS*NN
#define D_MODEL 256
#define HD2     512        // HEADS*PH
#define NHEADS  2
#define DKV     64
#define NN_     1024
#define BSZ     16

typedef __attribute__((ext_vector_type(2))) float v2f;
typedef __attribute__((ext_vector_type(8))) float v8f;

#define WMMA_F32(av, bv, acc) \
  __builtin_amdgcn_wmma_f32_16x16x4_f32(false, (av), false, (bv), (short)0, (acc), false, false)

// ---- ordered-float helpers for atomic max on floats --------------------
__device__ __forceinline__ unsigned f2ord(float f) {
  unsigned b = __float_as_uint(f);
  return (b & 0x80000000u) ? ~b : (b | 0x80000000u);
}
__device__ __forceinline__ float ord2f(unsigned k) {
  unsigned b = (k & 0x80000000u) ? (k & 0x7fffffffu) : ~k;
  return __uint_as_float(b);
}

// ========================================================================
// fp32 WMMA GEMM, register-blocked 16(M)x64(N) per wave, software-pipelined:
// next K-step fragments are loaded into fresh registers before the current
// 4 WMMAs issue, so s_wait_loadcnt can be partial and VMEM overlaps the
// matrix pipe. A fragment (2 contiguous floats) loads as one b64.
// ========================================================================
__global__ __launch_bounds__(128)
void gemm_nn_kernel(const float* __restrict__ A, const float* __restrict__ B,
                    float* __restrict__ C, int M, int Nc, int K,
                    int lda, int ldb, int ldc,
                    const float* __restrict__ bias1,
                    const float* __restrict__ bias2)
{
  const int wave = threadIdx.x >> 5;
  const int lane = threadIdx.x & 31;
  const int tm = (blockIdx.y * 4 + wave) * 16;
  const int tn = blockIdx.x * 64;
  if (tm >= M || tn >= Nc) return;
  const int half = lane >> 4;
  const int l15  = lane & 15;
  const int kk   = 2 * half;
  const float* Arow = A + (size_t)(tm + l15) * lda;
  const float* Bp   = B + tn + l15;
  v8f acc0 = {0.f,0.f,0.f,0.f,0.f,0.f,0.f,0.f};
  v8f acc1 = acc0, acc2 = acc0, acc3 = acc0;

  v2f av, b0v, b1v, b2v, b3v;
  {
    av = *(const v2f*)(Arow + kk);
    const float* b0 = Bp + (size_t)kk * ldb;
    const float* b1 = b0 + ldb;
    b0v.x = b0[0];  b0v.y = b1[0];
    b1v.x = b0[16]; b1v.y = b1[16];
    b2v.x = b0[32]; b2v.y = b1[32];
    b3v.x = b0[48]; b3v.y = b1[48];
  }
  for (int k = 4; k < K; k += 4) {
    v2f avn = *(const v2f*)(Arow + k + kk);
    const float* b0 = Bp + (size_t)(k + kk) * ldb;
    const float* b1 = b0 + ldb;
    v2f b0n, b1n, b2n, b3n;
    b0n.x = b0[0];  b0n.y = b1[0];
    b1n.x = b0[16]; b1n.y = b1[16];
    b2n.x = b0[32]; b2n.y = b1[32];
    b3n.x = b0[48]; b3n.y = b1[48];
    acc0 = WMMA_F32(av, b0v, acc0);
    acc1 = WMMA_F32(av, b1v, acc1);
    acc2 = WMMA_F32(av, b2v, acc2);
    acc3 = WMMA_F32(av, b3v, acc3);
    av = avn; b0v = b0n; b1v = b1n; b2v = b2n; b3v = b3n;
  }
  acc0 = WMMA_F32(av, b0v, acc0);
  acc1 = WMMA_F32(av, b1v, acc1);
  acc2 = WMMA_F32(av, b2v, acc2);
  acc3 = WMMA_F32(av, b3v, acc3);

  float bd0 = 0.f, bd1 = 0.f, bd2 = 0.f, bd3 = 0.f;
  if (bias1) {
    bd0 += bias1[tn + l15];      bd1 += bias1[tn + 16 + l15];
    bd2 += bias1[tn + 32 + l15]; bd3 += bias1[tn + 48 + l15];
  }
  if (bias2) {
    bd0 += bias2[tn + l15];      bd1 += bias2[tn + 16 + l15];
    bd2 += bias2[tn + 32 + l15]; bd3 += bias2[tn + 48 + l15];
  }
#pragma unroll
  for (int r = 0; r < 8; ++r) {
    float* crow = C + (size_t)(tm + r + 8*half) * ldc + tn + l15;
    crow[0]  = acc0[r] + bd0;
    crow[16] = acc1[r] + bd1;
    crow[32] = acc2[r] + bd2;
    crow[48] = acc3[r] + bd3;
  }
}

// ---- S[b,h] = (Q[b,h]/8) @ K[b,h]^T  (NT, 16x64/wave, all-b64 loads) ----
__global__ __launch_bounds__(128)
void attn_score_kernel(const float* __restrict__ Qp, const float* __restrict__ Kp,
                       float* __restrict__ S)
{
  const int wave = threadIdx.x >> 5;
  const int lane = threadIdx.x & 31;
  const int tm = (blockIdx.y * 4 + wave) * 16;   // query tile
  const int tn = blockIdx.x * 64;                // key tile base
  const int z  = blockIdx.z;                     // b*2+h
  const int b = z >> 1, h = z & 1;
  const float* Qb = Qp + (size_t)b * NN_ * (NHEADS*DKV) + h * DKV;
  const float* Kb = Kp + (size_t)b * NN_ * (NHEADS*DKV) + h * DKV;
  float* Sb = S + (size_t)z * NN_ * NN_;
  const int half = lane >> 4, l15 = lane & 15;
  const int kk = 2 * half;
  const float* qrow = Qb + (size_t)(tm + l15) * (NHEADS*DKV) + kk;
  const float* kr0  = Kb + (size_t)(tn +      l15) * (NHEADS*DKV) + kk;
  const float* kr1  = Kb + (size_t)(tn + 16 + l15) * (NHEADS*DKV) + kk;
  const float* kr2  = Kb + (size_t)(tn + 32 + l15) * (NHEADS*DKV) + kk;
  const float* kr3  = Kb + (size_t)(tn + 48 + l15) * (NHEADS*DKV) + kk;
  v8f acc0 = {0.f,0.f,0.f,0.f,0.f,0.f,0.f,0.f};
  v8f acc1 = acc0, acc2 = acc0, acc3 = acc0;

  v2f av  = *(const v2f*)(qrow);
  v2f b0v = *(const v2f*)(kr0);
  v2f b1v = *(const v2f*)(kr1);
  v2f b2v = *(const v2f*)(kr2);
  v2f b3v = *(const v2f*)(kr3);
  for (int k = 4; k < DKV; k += 4) {
    v2f avn = *(const v2f*)(qrow + k);
    v2f b0n = *(const v2f*)(kr0 + k);
    v2f b1n = *(const v2f*)(kr1 + k);
    v2f b2n = *(const v2f*)(kr2 + k);
    v2f b3n = *(const v2f*)(kr3 + k);
    acc0 = WMMA_F32(av, b0v, acc0);
    acc1 = WMMA_F32(av, b1v, acc1);
    acc2 = WMMA_F32(av, b2v, acc2);
    acc3 = WMMA_F32(av, b3v, acc3);
    av = avn; b0v = b0n; b1v = b1n; b2v = b2n; b3v = b3n;
  }
  acc0 = WMMA_F32(av, b0v, acc0);
  acc1 = WMMA_F32(av, b1v, acc1);
  acc2 = WMMA_F32(av, b2v, acc2);
  acc3 = WMMA_F32(av, b3v, acc3);

#pragma unroll
  for (int r = 0; r < 8; ++r) {
    float* srow = Sb + (size_t)(tm + r + 8*half) * NN_ + tn + l15;
    srow[0]  = acc0[r] * 0.125f;
    srow[16] = acc1[r] * 0.125f;
    srow[32] = acc2[r] * 0.125f;
    srow[48] = acc3[r] * 0.125f;
  }
}

// ---- O[b,h] = attn[b,h] @ V[b,h]  (NN, 16x64/wave covers all DKV) -------
__global__ __launch_bounds__(128)
void attn_out_kernel(const float* __restrict__ S, const float* __restrict__ Vp,
                     float* __restrict__ O)
{
  const int wave = threadIdx.x >> 5;
  const int lane = threadIdx.x & 31;
  const int tm = (blockIdx.y * 4 + wave) * 16;
  const int z  = blockIdx.z;
  const int b = z >> 1, h = z & 1;
  const float* Sb = S + (size_t)z * NN_ * NN_;
  const float* Vb = Vp + (size_t)b * NN_ * (NHEADS*DKV) + h * DKV;
  float* Ob = O + (size_t)b * NN_ * (NHEADS*DKV) + h * DKV;
  const int half = lane >> 4, l15 = lane & 15;
  const int kk = 2 * half;
  const float* srow = Sb + (size_t)(tm + l15) * NN_ + kk;
  const float* Vpl  = Vb + l15;
  v8f acc0 = {0.f,0.f,0.f,0.f,0.f,0.f,0.f,0.f};
  v8f acc1 = acc0, acc2 = acc0, acc3 = acc0;

  v2f av, b0v, b1v, b2v, b3v;
  {
    av = *(const v2f*)(srow);
    const float* v0 = Vpl + (size_t)kk * (NHEADS*DKV);
    const float* v1 = v0 + (NHEADS*DKV);
    b0v.x = v0[0];  b0v.y = v1[0];
    b1v.x = v0[16]; b1v.y = v1[16];
    b2v.x = v0[32]; b2v.y = v1[32];
    b3v.x = v0[48]; b3v.y = v1[48];
  }
  for (int k = 4; k < NN_; k += 4) {
    v2f avn = *(const v2f*)(srow + k);
    const float* v0 = Vpl + (size_t)(k + kk) * (NHEADS*DKV);
    const float* v1 = v0 + (NHEADS*DKV);
    v2f b0n, b1n, b2n, b3n;
    b0n.x = v0[0];  b0n.y = v1[0];
    b1n.x = v0[16]; b1n.y = v1[16];
    b2n.x = v0[32]; b2n.y = v1[32];
    b3n.x = v0[48]; b3n.y = v1[48];
    acc0 = WMMA_F32(av, b0v, acc0);
    acc1 = WMMA_F32(av, b1v, acc1);
    acc2 = WMMA_F32(av, b2v, acc2);
    acc3 = WMMA_F32(av, b3v, acc3);
    av = avn; b0v = b0n; b1v = b1n; b2v = b2n; b3v = b3n;
  }
  acc0 = WMMA_F32(av, b0v, acc0);
  acc1 = WMMA_F32(av, b1v, acc1);
  acc2 = WMMA_F32(av, b2v, acc2);
  acc3 = WMMA_F32(av, b3v, acc3);

#pragma unroll
  for (int r = 0; r < 8; ++r) {
    float* orow = Ob + (size_t)(tm + r + 8*half) * (NHEADS*DKV) + l15;
    orow[0]  = acc0[r];
    orow[16] = acc1[r];
    orow[32] = acc2[r];
    orow[48] = acc3[r];
  }
}

// ---- per-(node,head) alpha projections: one wave reduces C elems --------
__global__ __launch_bounds__(128)
void alpha_kernel(const float* __restrict__ Hm, const float* __restrict__ asrc,
                  const float* __restrict__ adst,
                  float* __restrict__ as_, float* __restrict__ ad_, int C)
{
  const int gw = (int)((blockIdx.x * (unsigned)blockDim.x + threadIdx.x) >> 5);
  const int lane = threadIdx.x & 31;
  if (gw >= N_TOT * NHEADS) return;
  const int n = gw >> 1, h = gw & 1;
  const float* hp = Hm + (size_t)n * (NHEADS * C) + h * C;
  const float* sp = asrc + h * C;
  const float* dp = adst + h * C;
  float s1 = 0.f, s2 = 0.f;
  for (int c = lane; c < C; c += 32) {
    float hv = hp[c];
    s1 += hv * sp[c];
    s2 += hv * dp[c];
  }
#pragma unroll
  for (int o = 16; o > 0; o >>= 1) {
    s1 += __shfl_down(s1, o, 32);
    s2 += __shfl_down(s2, o, 32);
  }
  if (lane == 0) { as_[gw] = s1; ad_[gw] = s2; }
}

__global__ void gat_init_kernel(unsigned* __restrict__ amax,
                                float* __restrict__ den, int n)
{
  int i = blockIdx.x * blockDim.x + threadIdx.x;
  if (i < n) { amax[i] = f2ord(-3.0e38f); den[i] = 0.f; }
}

__global__ void edge_max_kernel(const int* __restrict__ ei, int E0,
                                const float* __restrict__ as_,
                                const float* __restrict__ ad_,
                                unsigned* __restrict__ amax,
                                float* __restrict__ ae)
{
  int t = blockIdx.x * blockDim.x + threadIdx.x;
  int tot = (E0 + N_TOT) * NHEADS;
  if (t >= tot) return;
  int h = t & 1, e = t >> 1;
  int s = (e < E0) ? ei[e]      : (e - E0);
  int d = (e < E0) ? ei[E0 + e] : (e - E0);
  float a = as_[s * 2 + h] + ad_[d * 2 + h];
  a = (a > 0.f) ? a : 0.2f * a;          // leaky_relu 0.2
  ae[t] = a;
  atomicMax(&amax[d * 2 + h], f2ord(a));
}

__global__ void edge_expsum_kernel(const int* __restrict__ ei, int E0,
                                   const unsigned* __restrict__ amax,
                                   float* __restrict__ ae,
                                   float* __restrict__ den)
{
  int t = blockIdx.x * blockDim.x + threadIdx.x;
  int tot = (E0 + N_TOT) * NHEADS;
  if (t >= tot) return;
  int h = t & 1, e = t >> 1;
  int d = (e < E0) ? ei[E0 + e] : (e - E0);
  float ex = expf(ae[t] - ord2f(amax[d * 2 + h]));
  ae[t] = ex;
  atomicAdd(&den[d * 2 + h], ex);
}

// scatter-add h[src]*w into acc (preloaded with lin-branch + biases)
__global__ void edge_scatter_kernel(const int* __restrict__ ei, int E0,
                                    const float* __restrict__ Hm,
                                    const float* __restrict__ ae,
                                    const float* __restrict__ den,
                                    float* __restrict__ acc,
                                    int C, int concat, float headScale)
{
  const long long total  = (long long)(E0 + N_TOT) * NHEADS * C;
  const long long stride = (long long)gridDim.x * blockDim.x;
  for (long long t = (long long)blockIdx.x * blockDim.x + threadIdx.x;
       t < total; t += stride) {
    int c = (int)(t % C);
    long long eh = t / C;
    int h = (int)(eh & 1);
    int e = (int)(eh >> 1);
    int s = (e < E0) ? ei[e]      : (e - E0);
    int d = (e < E0) ? ei[E0 + e] : (e - E0);
    float w = ae[(size_t)e * 2 + h] / (den[(size_t)d * 2 + h] + 1e-16f);
    float val = Hm[(size_t)s * (NHEADS * C) + h * C + c] * w * headScale;
    float* dst = concat ? &acc[(size_t)d * (NHEADS * C) + h * C + c]
                        : &acc[(size_t)d * C + c];
    atomicAdd(dst, val);
  }
}

__global__ void elu_kernel(float* __restrict__ x, int n)
{
  int i = blockIdx.x * blockDim.x + threadIdx.x;
  if (i < n) { float v = x[i]; x[i] = (v > 0.f) ? v : (expf(v) - 1.f); }
}

__global__ void cat_kernel(const float* __restrict__ xg,
                           const float* __restrict__ o,
                           float* __restrict__ cat)
{
  int i = blockIdx.x * blockDim.x + threadIdx.x;
  if (i >= N_TOT * 256) return;
  int n = i >> 8, c = i & 255;
  cat[i] = (c < 128) ? xg[(size_t)n * 128 + c] : o[(size_t)n * 128 + (c - 128)];
}

// row softmax over 1024, in place
__global__ __launch_bounds__(256)
void softmax_kernel(float* __restrict__ S)
{
  __shared__ float sm[256];
  const int tid = threadIdx.x;
  float* p = S + (size_t)blockIdx.x * NN_;
  float m = -3.4e38f;
  for (int j = tid; j < NN_; j += 256) m = fmaxf(m, p[j]);
  sm[tid] = m; __syncthreads();
  for (int s = 128; s > 0; s >>= 1) {
    if (tid < s) sm[tid] = fmaxf(sm[tid], sm[tid + s]);
    __syncthreads();
  }
  m = sm[0]; __syncthreads();
  float sum = 0.f;
  for (int j = tid; j < NN_; j += 256) { float e = expf(p[j] - m); p[j] = e; sum += e; }
  sm[tid] = sum; __syncthreads();
  for (int s = 128; s > 0; s >>= 1) {
    if (tid < s) sm[tid] += sm[tid + s];
    __syncthreads();
  }
  float inv = 1.f / sm[0];
  for (int j = tid; j < NN_; j += 256) p[j] *= inv;
}

// y = LN(ybuf + q) * g + b, row length 256
__global__ __launch_bounds__(256)
void ln_kernel(const float* __restrict__ yb, const float* __restrict__ q,
               const float* __restrict__ g, const float* __restrict__ b,
               float* __restrict__ out)
{
  __shared__ float sm[256];
  const int tid = threadIdx.x;
  const size_t base = (size_t)blockIdx.x * 256;
  float v = yb[base + tid] + q[base + tid];
  sm[tid] = v; __syncthreads();
  for (int s = 128; s > 0; s >>= 1) { if (tid < s) sm[tid] += sm[tid + s]; __syncthreads(); }
  float mu = sm[0] * (1.f / 256.f); __syncthreads();
  float d = v - mu;
  sm[tid] = d * d; __syncthreads();
  for (int s = 128; s > 0; s >>= 1) { if (tid < s) sm[tid] += sm[tid + s]; __syncthreads(); }
  float var = sm[0] * (1.f / 256.f);
  out[base + tid] = d * rsqrtf(var + 1e-6f) * g[tid] + b[tid];
}

// ========================================================================
extern "C" void kernel_launch(void* const* d_in, const int* in_sizes, int n_in,
                              void* d_out, int out_size, void* d_ws, size_t ws_size,
                              hipStream_t stream)
{
  (void)n_in; (void)out_size; (void)ws_size;
  const float* q   = (const float*)d_in[0];
  const float* k   = (const float*)d_in[1];
  const float* v   = (const float*)d_in[2];
  const int*   ei  = (const int*)d_in[3];
  const float* Wq  = (const float*)d_in[4];
  const float* Wk  = (const float*)d_in[5];
  const float* Wv  = (const float*)d_in[6];
  const float* Wfc = (const float*)d_in[7];
  const float* g1W = (const float*)d_in[8];
  const float* g1as= (const float*)d_in[9];
  const float* g1ad= (const float*)d_in[10];
  const float* g1b = (const float*)d_in[11];
  const float* l1W = (const float*)d_in[12];
  const float* l1b = (const float*)d_in[13];
  const float* g2W = (const float*)d_in[14];
  const float* g2as= (const float*)d_in[15];
  const float* g2ad= (const float*)d_in[16];
  const float* g2b = (const float*)d_in[17];
  const float* l2W = (const float*)d_in[18];
  const float* l2b = (const float*)d_in[19];
  const float* g3W = (const float*)d_in[20];
  const float* g3as= (const float*)d_in[21];
  const float* g3ad= (const float*)d_in[22];
  const float* g3b = (const float*)d_in[23];
  const float* l3W = (const float*)d_in[24];
  const float* l3b = (const float*)d_in[25];
  const float* lng = (const float*)d_in[26];
  const float* lnb = (const float*)d_in[27];

  const int E0   = in_sizes[3] / 2;      // 131072 graph edges
  const int Etot = E0 + N_TOT;           // + self loops

  // workspace carve-up (floats)
  float* ws = (float*)d_ws;
  size_t off = 0;
  float* bufH = ws + off; off += (size_t)N_TOT * HD2;   // GAT h matrices / cat+ybuf
  float* bufA = ws + off; off += (size_t)N_TOT * HD2;   // x1 / layer3 acc (xg)
  float* bufB = ws + off; off += (size_t)N_TOT * HD2;   // x2 / Q,K,V,O
  float* as_  = ws + off; off += (size_t)N_TOT * NHEADS;
  float* ad_  = ws + off; off += (size_t)N_TOT * NHEADS;
  unsigned* amax = (unsigned*)(ws + off); off += (size_t)N_TOT * NHEADS;
  float* den  = ws + off; off += (size_t)N_TOT * NHEADS;
  float* ae   = ws + off; off += (size_t)Etot * NHEADS;

  const float* npf = nullptr;
  const dim3 blk128(128), blk256(256);
  const int nhBlocks   = (N_TOT * NHEADS + 255) / 256;        // 128
  const int edgeBlocks = (Etot * NHEADS + 255) / 256;         // 1152
  const int alphaBlks  = (N_TOT * NHEADS) / 4;                // 4 waves/block
  const int Mb = N_TOT / 64;                                  // 256 row-blocks

  // ===== GAT layer 1 (C=256, concat) =====
  gemm_nn_kernel<<<dim3(HD2/64, Mb), blk128, 0, stream>>>(
      q, g1W, bufH, N_TOT, HD2, D_MODEL, D_MODEL, HD2, HD2, npf, npf);
  gemm_nn_kernel<<<dim3(HD2/64, Mb), blk128, 0, stream>>>(
      q, l1W, bufA, N_TOT, HD2, D_MODEL, D_MODEL, HD2, HD2, l1b, g1b);
  alpha_kernel<<<alphaBlks, blk128, 0, stream>>>(bufH, g1as, g1ad, as_, ad_, 256);
  gat_init_kernel<<<nhBlocks, blk256, 0, stream>>>(amax, den, N_TOT * NHEADS);
  edge_max_kernel<<<edgeBlocks, blk256, 0, stream>>>(ei, E0, as_, ad_, amax, ae);
  edge_expsum_kernel<<<edgeBlocks, blk256, 0, stream>>>(ei, E0, amax, ae, den);
  edge_scatter_kernel<<<8192, blk256, 0, stream>>>(ei, E0, bufH, ae, den, bufA,
                                                   256, 1, 1.0f);
  elu_kernel<<<(N_TOT * HD2) / 256, blk256, 0, stream>>>(bufA, N_TOT * HD2);

  // ===== GAT layer 2 (C=256, concat), x = bufA =====
  gemm_nn_kernel<<<dim3(HD2/64, Mb), blk128, 0, stream>>>(
      bufA, g2W, bufH, N_TOT, HD2, HD2, HD2, HD2, HD2, npf, npf);
  gemm_nn_kernel<<<dim3(HD2/64, Mb), blk128, 0, stream>>>(
      bufA, l2W, bufB, N_TOT, HD2, HD2, HD2, HD2, HD2, l2b, g2b);
  alpha_kernel<<<alphaBlks, blk128, 0, stream>>>(bufH, g2as, g2ad, as_, ad_, 256);
  gat_init_kernel<<<nhBlocks, blk256, 0, stream>>>(amax, den, N_TOT * NHEADS);
  edge_max_kernel<<<edgeBlocks, blk256, 0, stream>>>(ei, E0, as_, ad_, amax, ae);
  edge_expsum_kernel<<<edgeBlocks, blk256, 0, stream>>>(ei, E0, amax, ae, den);
  edge_scatter_kernel<<<8192, blk256, 0, stream>>>(ei, E0, bufH, ae, den, bufB,
                                                   256, 1, 1.0f);
  elu_kernel<<<(N_TOT * HD2) / 256, blk256, 0, stream>>>(bufB, N_TOT * HD2);

  // ===== GAT layer 3 (C=128, head mean), x = bufB, out (xg) = bufA[N,128] =====
  gemm_nn_kernel<<<dim3(256/64, Mb), blk128, 0, stream>>>(
      bufB, g3W, bufH, N_TOT, 256, HD2, HD2, 256, 256, npf, npf);
  gemm_nn_kernel<<<dim3(128/64, Mb), blk128, 0, stream>>>(
      bufB, l3W, bufA, N_TOT, 128, HD2, HD2, 128, 128, l3b, g3b);
  alpha_kernel<<<alphaBlks, blk128, 0, stream>>>(bufH, g3as, g3ad, as_, ad_, 128);
  gat_init_kernel<<<nhBlocks, blk256, 0, stream>>>(amax, den, N_TOT * NHEADS);
  edge_max_kernel<<<edgeBlocks, blk256, 0, stream>>>(ei, E0, as_, ad_, amax, ae);
  edge_expsum_kernel<<<edgeBlocks, blk256, 0, stream>>>(ei, E0, amax, ae, den);
  edge_scatter_kernel<<<8192, blk256, 0, stream>>>(ei, E0, bufH, ae, den, bufA,
                                                   128, 0, 0.5f);   // mean over 2 heads

  // ===== Q/K/V projections (bufB free after layer-3 GEMMs issued) =====
  float* Qp = bufB;
  float* Kp = bufB + (size_t)N_TOT * 128;
  float* Vp = bufB + (size_t)N_TOT * 256;
  float* Op = bufB + (size_t)N_TOT * 384;
  gemm_nn_kernel<<<dim3(128/64, Mb), blk128, 0, stream>>>(
      q, Wq, Qp, N_TOT, 128, D_MODEL, D_MODEL, 128, 128, npf, npf);
  gemm_nn_kernel<<<dim3(128/64, Mb), blk128, 0, stream>>>(
      k, Wk, Kp, N_TOT, 128, D_MODEL, D_MODEL, 128, 128, npf, npf);
  gemm_nn_kernel<<<dim3(128/64, Mb), blk128, 0, stream>>>(
      v, Wv, Vp, N_TOT, 128, D_MODEL, D_MODEL, 128, 128, npf, npf);

  // ===== attention: scores + softmax (in place in d_out) + output =====
  float* attn = (float*)d_out + (size_t)N_TOT * D_MODEL;
  attn_score_kernel<<<dim3(NN_/64, NN_/64, BSZ*NHEADS), blk128, 0, stream>>>(Qp, Kp, attn);
  softmax_kernel<<<BSZ * NHEADS * NN_, blk256, 0, stream>>>(attn);
  attn_out_kernel<<<dim3(1, NN_/64, BSZ*NHEADS), blk128, 0, stream>>>(attn, Vp, Op);

  // ===== concat + Wfc + residual LayerNorm =====
  float* catb = bufH;                         // [N,256]
  float* ybuf = bufH + (size_t)N_TOT * 256;   // [N,256]
  cat_kernel<<<(N_TOT * 256) / 256, blk256, 0, stream>>>(bufA, Op, catb);
  gemm_nn_kernel<<<dim3(256/64, Mb), blk128, 0, stream>>>(
      catb, Wfc, ybuf, N_TOT, 256, 256, 256, 256, 256, npf, npf);
  ln_kernel<<<N_TOT, blk256, 0, stream>>>(ybuf, q, lng, lnb, (float*)d_out);
}